// NoisyTopItemsPerExpertRouter_90263032692928
// MI455X (gfx1250) — compile-verified
//
#include <hip/hip_runtime.h>
#include <hip/hip_bf16.h>
#include <math.h>

// ---------------- problem constants (fixed by reference) ----------------
constexpr int T_TOK = 16384;   // B*S
constexpr int HDIM  = 4096;
constexpr int NEXP  = 64;
constexpr int CSEL  = 512;
constexpr float EPSV = 1e-6f;

typedef __attribute__((ext_vector_type(2))) float v2f;
typedef __attribute__((ext_vector_type(8))) float v8f;

// =======================================================================
// Kernel 1: logits[E,T] = gate_w[E,H] x hidden[T,H]^T  via fp32 WMMA
//   grid: 256 blocks (64 tokens each), 512 threads = 16 waves
//   each wave owns one 16(expert) x 16(token) tile; K staged in LDS chunks
// =======================================================================
__global__ __launch_bounds__(512)
void gemm_logits_kernel(const float* __restrict__ hidden,
                        const float* __restrict__ gate,
                        float* __restrict__ logits) {
    constexpr int KCH = 32;
    constexpr int LDR = 34;                 // padded row stride (even -> 8B aligned pairs)
    __shared__ float lA[NEXP * LDR];        // gate chunk   [expert][k]
    __shared__ float lB[64 * LDR];          // hidden chunk [token ][k]

    const int tid   = threadIdx.x;
    const int wave  = tid >> 5;
    const int lane  = tid & 31;
    const int lo    = lane & 15;
    const int hi    = lane >> 4;            // 0 or 1
    const int t_blk = blockIdx.x * 64;

    const int e0  = (wave >> 2) * 16;       // expert tile base (0..48)
    const int t0w = (wave & 3) * 16;        // token  tile base within block

    // cooperative loader mapping: 64 rows x 32 k, float4 per thread
    const int lrow = tid >> 3;              // 0..63
    const int lcol = (tid & 7) * 4;         // 0..28

    v8f acc = {};

    for (int k0 = 0; k0 < HDIM; k0 += KCH) {
        const float4 ga = *(const float4*)(gate  + (size_t)lrow * HDIM + k0 + lcol);
        const float4 hb = *(const float4*)(hidden + (size_t)(t_blk + lrow) * HDIM + k0 + lcol);
        if (k0 + KCH < HDIM)  // hint the streaming read of the next chunk
            __builtin_prefetch(hidden + (size_t)(t_blk + lrow) * HDIM + k0 + KCH + lcol, 0, 0);

        __syncthreads();      // previous chunk fully consumed
        lA[lrow * LDR + lcol + 0] = ga.x;
        lA[lrow * LDR + lcol + 1] = ga.y;
        lA[lrow * LDR + lcol + 2] = ga.z;
        lA[lrow * LDR + lcol + 3] = ga.w;
        lB[lrow * LDR + lcol + 0] = hb.x;
        lB[lrow * LDR + lcol + 1] = hb.y;
        lB[lrow * LDR + lcol + 2] = hb.z;
        lB[lrow * LDR + lcol + 3] = hb.w;
        __syncthreads();

#pragma unroll
        for (int kk = 0; kk < KCH; kk += 4) {
            const int ks = kk + hi * 2;     // lanes 0-15: K=0,1 ; lanes 16-31: K=2,3
            v2f a, b;
            a.x = lA[(e0 + lo) * LDR + ks];
            a.y = lA[(e0 + lo) * LDR + ks + 1];
            b.x = lB[(t0w + lo) * LDR + ks];
            b.y = lB[(t0w + lo) * LDR + ks + 1];
            // D = A(16x4,f32) * B(4x16,f32) + C ; 8 args like the f16 variant
            acc = __builtin_amdgcn_wmma_f32_16x16x4_f32(
                false, a, false, b, (short)0, acc, false, false);
        }
    }

    // C/D layout: VGPR r -> M = r (lanes 0-15) / r+8 (lanes 16-31), N = lane&15
#pragma unroll
    for (int r = 0; r < 8; ++r) {
        const int e = e0 + r + hi * 8;
        const int t = t_blk + t0w + lo;
        logits[(size_t)e * T_TOK + t] = acc[r];
    }
}

// monotone float -> uint32 key (order-preserving)
__device__ __forceinline__ unsigned f2key(float f) {
    unsigned u = __float_as_uint(f);
    return (u & 0x80000000u) ? ~u : (u | 0x80000000u);
}

// =======================================================================
// Kernel 2: per-expert top-512 via 4-pass 8-bit MSB radix select.
//   grid: 64 blocks (one expert), 256 threads. Row is L2-resident.
//   Emits 512-word bitmask per expert.
// =======================================================================
__global__ __launch_bounds__(256)
void topk_select_kernel(const float* __restrict__ logits,
                        unsigned* __restrict__ masks) {
    const int e   = blockIdx.x;
    const int tid = threadIdx.x;
    const float* row = logits + (size_t)e * T_TOK;

    __shared__ unsigned hist[256];
    __shared__ unsigned mw[T_TOK / 32];     // 512 words
    __shared__ unsigned sBin, sK, sEq;

    unsigned prefixKey = 0, prefixMask = 0;
    unsigned k = CSEL;

    for (int pass = 0; pass < 4; ++pass) {
        const int shift = 24 - pass * 8;
        hist[tid] = 0;
        __syncthreads();
        for (int t = tid; t < T_TOK; t += 256) {
            unsigned key = f2key(row[t]);
            if ((key & prefixMask) == prefixKey)
                atomicAdd(&hist[(key >> shift) & 255u], 1u);
        }
        __syncthreads();
        if (tid == 0) {
            unsigned cum = 0;
            int b = 255;
            for (; b > 0; --b) {
                unsigned c = hist[b];
                if (cum + c >= k) break;
                cum += c;
            }
            sBin = (unsigned)b;
            sK   = k - cum;                 // rank within the chosen bin
        }
        __syncthreads();
        prefixKey |= (sBin << shift);
        prefixMask |= (255u << shift);
        k = sK;
        __syncthreads();
    }

    const unsigned thrKey = prefixKey;      // exact key of the C-th largest
    const unsigned kEq    = k;              // #ties at threshold to include (>=1)

    if (tid == 0) sEq = 0;
    for (int w = tid; w < T_TOK / 32; w += 256) mw[w] = 0;
    __syncthreads();

    for (int t = tid; t < T_TOK; t += 256) {
        unsigned key = f2key(row[t]);
        bool sel = key > thrKey;
        if (!sel && key == thrKey) sel = (atomicAdd(&sEq, 1u) < kEq);
        if (sel) atomicOr(&mw[t >> 5], 1u << (t & 31));
    }
    __syncthreads();
    for (int w = tid; w < T_TOK / 32; w += 256)
        masks[e * (T_TOK / 32) + w] = mw[w];
}

// =======================================================================
// Kernel 3: fused softmax + mask + renormalize + importance partials.
//   grid: 256 blocks (64 tokens each), 256 threads.
// =======================================================================
__global__ __launch_bounds__(256)
void finalize_kernel(const float* __restrict__ logits,
                     const unsigned* __restrict__ masks,
                     float* __restrict__ probs_out,
                     float* __restrict__ routing_out,
                     float* __restrict__ importance) {
    constexpr int LDP = NEXP + 1;           // padded
    __shared__ float lt[64 * LDP];          // [token][expert]
    __shared__ unsigned lm[NEXP * 2];       // [expert][2 words covering 64 tokens]

    const int tid = threadIdx.x;
    const int t0  = blockIdx.x * 64;

    for (int idx = tid; idx < NEXP * 64; idx += 256) {
        const int e  = idx >> 6;
        const int tt = idx & 63;
        lt[tt * LDP + e] = logits[(size_t)e * T_TOK + t0 + tt];
    }
    if (tid < NEXP * 2) {
        const int e = tid >> 1, w = tid & 1;
        lm[tid] = masks[e * (T_TOK / 32) + (t0 >> 5) + w];
    }
    __syncthreads();

    if (tid < 64) {
        const int tt = tid;
        float m = -INFINITY;
        for (int e = 0; e < NEXP; ++e) m = fmaxf(m, lt[tt * LDP + e]);
        float s = 0.f;
        for (int e = 0; e < NEXP; ++e) s += __expf(lt[tt * LDP + e] - m);
        const float inv_s = 1.f / s;
        float ms = 0.f;
        for (int e = 0; e < NEXP; ++e) {
            float p = __expf(lt[tt * LDP + e] - m) * inv_s;
            const unsigned sel = (lm[e * 2 + (tt >> 5)] >> (tt & 31)) & 1u;
            p *= (float)sel;
            ms += p;
            lt[tt * LDP + e] = p;           // overwrite with masked prob
        }
        const float inv = 1.f / (ms + EPSV);
        for (int e = 0; e < NEXP; ++e) lt[tt * LDP + e] *= inv;
    }
    __syncthreads();

    if (tid < NEXP) {                        // per-expert partial importance
        float sum = 0.f;
        for (int tt = 0; tt < 64; ++tt) sum += lt[tt * LDP + tid];
        atomicAdd(&importance[tid], sum);
    }

    for (int idx = tid; idx < NEXP * 64; idx += 256) {  // coalesced [t][e] writes
        const int tt = idx >> 6;
        const int e  = idx & 63;
        const unsigned sel = (lm[e * 2 + (tt >> 5)] >> (tt & 31)) & 1u;
        probs_out[(size_t)(t0 + tt) * NEXP + e]   = lt[tt * LDP + e];
        routing_out[(size_t)(t0 + tt) * NEXP + e] = (float)sel;
    }
}

__global__ void zero_importance_kernel(float* imp) { imp[threadIdx.x] = 0.f; }

// aux = imp_loss + load_loss; load[e] == C exactly -> std(load) == 0 -> load_loss == 0
__global__ void aux_loss_kernel(const float* __restrict__ importance,
                                float* __restrict__ out) {
    if (threadIdx.x == 0) {
        float mean = 0.f;
        for (int e = 0; e < NEXP; ++e) mean += importance[e];
        mean *= (1.f / NEXP);
        float var = 0.f;
        for (int e = 0; e < NEXP; ++e) {
            const float d = importance[e] - mean;
            var += d * d;
        }
        var *= (1.f / (NEXP - 1));          // unbiased (ddof=1)
        const float denom = mean + EPSV;
        out[0] = var / (denom * denom);
    }
}

// =======================================================================
extern "C" void kernel_launch(void* const* d_in, const int* in_sizes, int n_in,
                              void* d_out, int out_size, void* d_ws, size_t ws_size,
                              hipStream_t stream) {
    const float* hidden = (const float*)d_in[0];   // [T, H]
    const float* gate   = (const float*)d_in[1];   // [E, H]
    // d_in[2] is C == 512 (compile-time constant here)

    float*    logits     = (float*)d_ws;                           // E*T f32
    unsigned* masks      = (unsigned*)(logits + (size_t)NEXP * T_TOK); // E*512 u32
    float*    importance = (float*)(masks + NEXP * (T_TOK / 32));  // E f32

    float* probs_out   = (float*)d_out;                            // [T,E]
    float* routing_out = probs_out + (size_t)T_TOK * NEXP;         // [T,E]
    float* aux_out     = routing_out + (size_t)T_TOK * NEXP;       // [1]

    gemm_logits_kernel<<<T_TOK / 64, 512, 0, stream>>>(hidden, gate, logits);
    topk_select_kernel<<<NEXP, 256, 0, stream>>>(logits, masks);
    zero_importance_kernel<<<1, NEXP, 0, stream>>>(importance);
    finalize_kernel<<<T_TOK / 64, 256, 0, stream>>>(logits, masks, probs_out,
                                                    routing_out, importance);
    aux_loss_kernel<<<1, 32, 0, stream>>>(importance, aux_out);
}